// GridGenerator_14705968021845
// MI455X (gfx1250) — compile-verified
//
#include <hip/hip_runtime.h>

#define NUM_FID 20
#define IMG_H   48
#define IMG_W   160
#define BATCH   128
#define NPTS    30                 // 20 kernel terms + 10 polynomial terms
#define GPTS    (IMG_H * IMG_W)    // 7680 grid points
#define NCOLS   (BATCH * 2)        // 256 columns (batch x {x,y})

typedef float v2f __attribute__((ext_vector_type(2)));
typedef float v8f __attribute__((ext_vector_type(8)));

// ---------- constant geometry helpers ----------
__device__ __forceinline__ float ctrl_x(int n) {
    return -1.0f + 2.0f * (float)(n % 10) / 9.0f;   // linspace(-1,1,10)
}
__device__ __forceinline__ float ctrl_y(int n) {
    return (n < 10) ? -1.0f : 1.0f;
}
__device__ __forceinline__ float tps_k(float dx, float dy) {
    float r2 = dx * dx + dy * dy;
    r2 = fmaxf(r2, 1e-8f);                          // clip(r2, 1e-8, None)
    return r2 * (0.5f * logf(r2));
}
__device__ __forceinline__ float poly_term(float x, float y, int idx) {
    switch (idx) {
        case 0: return 1.0f;
        case 1: return x;
        case 2: return y;
        case 3: return x * x;
        case 4: return x * y;
        case 5: return y * y;
        case 6: return x * x * x;
        case 7: return x * x * y;
        case 8: return x * y * y;
        default: return y * y * y;
    }
}

// ---------- kernel 1: build L and invert (partial-pivot Gauss-Jordan) ----------
// Writes W[i*20 + n] = (L^-1)[i][n] for n < 20 (only first 20 cols matter:
// target rows 20..29 are zero).
__global__ void tps_invert_kernel(float* __restrict__ W) {
    __shared__ float A[NPTS][2 * NPTS];
    __shared__ float fac[NPTS];
    __shared__ int   piv;
    const int tid = threadIdx.x;

    // Build augmented [L | I]
    for (int t = tid; t < NPTS * 2 * NPTS; t += blockDim.x) {
        int r = t / (2 * NPTS);
        int j = t % (2 * NPTS);
        float v;
        if (j < NPTS) {
            if (r < NUM_FID && j < NUM_FID) {
                v = tps_k(ctrl_x(r) - ctrl_x(j), ctrl_y(r) - ctrl_y(j));
            } else if (r < NUM_FID) {               // P block
                v = poly_term(ctrl_x(r), ctrl_y(r), j - NUM_FID);
            } else if (j < NUM_FID) {               // P^T block
                v = poly_term(ctrl_x(j), ctrl_y(j), r - NUM_FID);
            } else {
                v = 0.0f;                            // zero block
            }
        } else {
            v = ((j - NPTS) == r) ? 1.0f : 0.0f;     // identity
        }
        A[r][j] = v;
    }
    __syncthreads();

    for (int k = 0; k < NPTS; ++k) {
        if (tid == 0) {
            int p = k; float best = fabsf(A[k][k]);
            for (int r = k + 1; r < NPTS; ++r) {
                float v = fabsf(A[r][k]);
                if (v > best) { best = v; p = r; }
            }
            piv = p;
        }
        __syncthreads();
        const int p = piv;                  // uniform
        if (p != k) {
            for (int j = tid; j < 2 * NPTS; j += blockDim.x) {
                float t = A[k][j]; A[k][j] = A[p][j]; A[p][j] = t;
            }
        }
        __syncthreads();
        const float pv = A[k][k];
        for (int j = tid; j < 2 * NPTS; j += blockDim.x) A[k][j] /= pv;
        __syncthreads();
        if (tid < NPTS) fac[tid] = A[tid][k];
        __syncthreads();
        for (int t = tid; t < NPTS * 2 * NPTS; t += blockDim.x) {
            int r = t / (2 * NPTS);
            int j = t % (2 * NPTS);
            if (r != k) A[r][j] -= fac[r] * A[k][j];
        }
        __syncthreads();
    }

    for (int t = tid; t < NPTS * NUM_FID; t += blockDim.x) {
        int i = t / NUM_FID, n = t % NUM_FID;
        W[t] = A[i][NPTS + n];
    }
}

// ---------- kernel 2: M[g,:] = phi(g) . W ; pack S from input ----------
__global__ void tps_basis_kernel(const float* __restrict__ scp,
                                 const float* __restrict__ W,
                                 float* __restrict__ Mmat,
                                 float* __restrict__ S) {
    __shared__ float Wl[NPTS * NUM_FID];
    for (int t = threadIdx.x; t < NPTS * NUM_FID; t += blockDim.x) Wl[t] = W[t];
    __syncthreads();

    const int g = blockIdx.x * blockDim.x + threadIdx.x;

    // Pack S[n*256 + 2b + c] = scp[b, n, c]
    if (g < NUM_FID * NCOLS) {
        int n = g >> 8, j = g & 255;
        int b = j >> 1, c = j & 1;
        S[g] = scp[b * NUM_FID * 2 + n * 2 + c];
    }

    if (g < GPTS) {
        const int iy = g / IMG_W, ix = g % IMG_W;
        const float x = -1.0f + 2.0f * (float)ix / (float)(IMG_W - 1);
        const float y = -1.0f + 2.0f * (float)iy / (float)(IMG_H - 1);

        float phi[NPTS];
        #pragma unroll
        for (int n = 0; n < NUM_FID; ++n)
            phi[n] = tps_k(x - ctrl_x(n), y - ctrl_y(n));
        phi[20] = 1.0f;  phi[21] = x;      phi[22] = y;
        phi[23] = x * x; phi[24] = x * y;  phi[25] = y * y;
        phi[26] = x * x * x; phi[27] = x * x * y;
        phi[28] = x * y * y; phi[29] = y * y * y;

        #pragma unroll 4
        for (int n = 0; n < NUM_FID; ++n) {
            float acc = 0.0f;
            #pragma unroll
            for (int i = 0; i < NPTS; ++i) acc += phi[i] * Wl[i * NUM_FID + n];
            Mmat[g * NUM_FID + n] = acc;
        }
    }
}

// ---------- kernel 3: Out(7680x256) = M(7680x20) . S(20x256) via f32 WMMA ----
// One wave per 16x16 output tile; K=20 as five 16x16x4 fp32 WMMA steps.
__global__ void tps_wmma_kernel(const float* __restrict__ Mmat,
                                const float* __restrict__ S,
                                float* __restrict__ out) {
    const int lane   = threadIdx.x & 31;
    const int wave   = blockIdx.x * (blockDim.x >> 5) + (threadIdx.x >> 5);
    const int tile_m = wave >> 4;     // 0..479  (grid-point tiles)
    const int tile_n = wave & 15;     // 0..15   (column tiles)

    const bool hi   = lane >= 16;     // lanes 16-31 carry K+2,K+3
    const int  l    = lane & 15;
    const int  koff = hi ? 2 : 0;

    const int arow = tile_m * 16 + l;     // A: lane -> matrix row M
    const int bcol = tile_n * 16 + l;     // B: lane -> matrix col N

    v8f acc = {0.f, 0.f, 0.f, 0.f, 0.f, 0.f, 0.f, 0.f};

    #pragma unroll
    for (int kk = 0; kk < NUM_FID; kk += 4) {
        v2f a, b;
        a.x = Mmat[arow * NUM_FID + kk + koff];
        a.y = Mmat[arow * NUM_FID + kk + koff + 1];
        b.x = S[(kk + koff)     * NCOLS + bcol];
        b.y = S[(kk + koff + 1) * NCOLS + bcol];
        // (neg_a, A, neg_b, B, c_mod, C, reuse_a, reuse_b)
        acc = __builtin_amdgcn_wmma_f32_16x16x4_f32(
            false, a, false, b, (short)0, acc, false, false);
    }

    // C/D layout: VGPR v holds row (v) for lanes 0-15 and row (v+8) for 16-31;
    // lane (mod 16) is the column. Scatter into (B, H, W, 2) layout.
    const int bb = bcol >> 1;
    const int cc = bcol & 1;
    const int rowbase = tile_m * 16 + (hi ? 8 : 0);
    #pragma unroll
    for (int v = 0; v < 8; ++v) {
        out[bb * (GPTS * 2) + (rowbase + v) * 2 + cc] = acc[v];
    }
}

extern "C" void kernel_launch(void* const* d_in, const int* in_sizes, int n_in,
                              void* d_out, int out_size, void* d_ws, size_t ws_size,
                              hipStream_t stream) {
    const float* scp = (const float*)d_in[0];   // (128, 20, 2) fp32
    float* out = (float*)d_out;                 // (128, 48, 160, 2) fp32

    // Workspace layout (floats): W[600] | S[5120] | M[153600]  ~= 623 KB
    float* W    = (float*)d_ws;
    float* S    = W + NPTS * NUM_FID;
    float* Mmat = S + NUM_FID * NCOLS;

    tps_invert_kernel<<<1, 64, 0, stream>>>(W);
    tps_basis_kernel<<<(GPTS + 255) / 256, 256, 0, stream>>>(scp, W, Mmat, S);

    const int tiles = (GPTS / 16) * (NCOLS / 16);   // 7680 tiles, 1 wave each
    tps_wmma_kernel<<<tiles / 8, 256, 0, stream>>>(Mmat, S, out);
}